// MWI_64991445123199
// MI455X (gfx1250) — compile-verified
//
#include <hip/hip_runtime.h>
#include <math.h>

typedef float v2f __attribute__((ext_vector_type(2)));
typedef float v8f __attribute__((ext_vector_type(8)));

#define B_DIM 4
#define C_DIM 256
#define K_DIM 64
#define HW    4096   // 64*64
#define W3C   768    // 3*C

static __device__ __forceinline__ v8f wmma_f32(v2f a, v2f b, v8f c) {
  // D = A(16x4 f32) * B(4x16 f32) + C(16x16 f32)
  return __builtin_amdgcn_wmma_f32_16x16x4_f32(false, a, false, b, (short)0, c,
                                               false, false);
}

static __device__ __forceinline__ v8f vzero8() {
  v8f z;
#pragma unroll
  for (int i = 0; i < 8; ++i) z[i] = 0.0f;
  return z;
}

// #{i in [0,s) : 0 <= t - i + p < 64}
static __device__ __forceinline__ int cnt1d(int t, int s, int p) {
  int lo = t + p - 63; if (lo < 0) lo = 0;
  int hi = t + p;      if (hi > s - 1) hi = s - 1;
  return hi - lo + 1;
}

// ---------------------------------------------------------------------------
// prep: xT[b][yx][c] = x[b][c][yx]; w_catT[i][o] = w_cat[o][i];
//       mu[b][c][k] = muT[b][k][c] = mu0[c][k]
// ---------------------------------------------------------------------------
__global__ void prep_kernel(const float* __restrict__ x,
                            const float* __restrict__ mu0,
                            const float* __restrict__ w_cat,
                            float* __restrict__ xT, float* __restrict__ w_catT,
                            float* __restrict__ mu, float* __restrict__ muT) {
  int tid = blockIdx.x * blockDim.x + threadIdx.x;
  if (tid < B_DIM * HW * C_DIM) {
    int c  = tid & 255;
    int yx = (tid >> 8) & 4095;
    int b  = tid >> 20;
    xT[tid] = x[((size_t)b * C_DIM + c) * HW + yx];
  }
  if (tid < W3C * C_DIM) {  // 196608
    int o = tid & 255;
    int i = tid >> 8;
    w_catT[i * C_DIM + o] = w_cat[o * W3C + i];
  }
  if (tid < B_DIM * C_DIM * K_DIM) {  // 65536
    int k = tid & 63;
    int c = (tid >> 6) & 255;
    int b = tid >> 14;
    float v = mu0[c * K_DIM + k];
    mu[tid] = v;
    muT[((size_t)b * K_DIM + k) * C_DIM + c] = v;
  }
}

// ---------------------------------------------------------------------------
// GEMM1 + softmax: logits P = xT(HW x C) * mu(C x K); zsw = cnt2d * softmax_k(P)
// one wave per 16-row strip, all 64 k-columns (4 WMMA tiles), contraction C=256
// ---------------------------------------------------------------------------
__global__ void gemm1_softmax_kernel(const float* __restrict__ xT,
                                     const float* __restrict__ mu,
                                     float* __restrict__ zsw, int s, int p) {
  const int lane = threadIdx.x & 31;
  const int l    = lane & 15;
  const int half = lane >> 4;
  const int b    = blockIdx.y;
  const int row0 = blockIdx.x * 16;
  const float* xTb = xT + (size_t)b * HW * C_DIM;
  const float* mub = mu + (size_t)b * C_DIM * K_DIM;

  v8f acc[4];
#pragma unroll
  for (int t = 0; t < 4; ++t) acc[t] = vzero8();

  const float* arow = xTb + (size_t)(row0 + l) * C_DIM + 2 * half;
  for (int c0 = 0; c0 < C_DIM; c0 += 4) {
    v2f a = *(const v2f*)(arow + c0);
    const float* bp = mub + (size_t)(c0 + 2 * half) * K_DIM + l;
#pragma unroll
    for (int t = 0; t < 4; ++t) {
      v2f bf;
      bf.x = bp[t * 16];
      bf.y = bp[K_DIM + t * 16];
      acc[t] = wmma_f32(a, bf, acc[t]);
    }
  }

  float* zb = zsw + (size_t)b * HW * K_DIM;
#pragma unroll
  for (int r = 0; r < 8; ++r) {
    int yx = row0 + r + 8 * half;
    float m = acc[0][r];
    m = fmaxf(m, acc[1][r]); m = fmaxf(m, acc[2][r]); m = fmaxf(m, acc[3][r]);
    for (int off = 8; off; off >>= 1) m = fmaxf(m, __shfl_xor(m, off, 16));
    float e[4]; float ssum = 0.0f;
#pragma unroll
    for (int t = 0; t < 4; ++t) { e[t] = expf(acc[t][r] - m); ssum += e[t]; }
    for (int off = 8; off; off >>= 1) ssum += __shfl_xor(ssum, off, 16);
    int Y = yx >> 6, X = yx & 63;
    float cw  = (float)(cnt1d(Y, s, p) * cnt1d(X, s, p));
    float inv = cw / ssum;
#pragma unroll
    for (int t = 0; t < 4; ++t)
      zb[(size_t)yx * K_DIM + t * 16 + l] = e[t] * inv;
  }
}

// ---------------------------------------------------------------------------
// colsum[b][k] = sum_yx zsw[b][yx][k] + OOB/K   (deterministic tree reduce)
// ---------------------------------------------------------------------------
__global__ void colsum_kernel(const float* __restrict__ zsw,
                              float* __restrict__ colsum, float oobterm) {
  __shared__ float sm[256];
  int k = blockIdx.x, b = blockIdx.y, t = threadIdx.x;
  const float* zb = zsw + (size_t)b * HW * K_DIM + k;
  float sum = 0.0f;
  for (int j = 0; j < 16; ++j) sum += zb[(size_t)(t + 256 * j) * K_DIM];
  sm[t] = sum; __syncthreads();
  for (int off = 128; off; off >>= 1) {
    if (t < off) sm[t] += sm[t + off];
    __syncthreads();
  }
  if (t == 0) colsum[b * K_DIM + k] = sm[0] + oobterm;
}

// ---------------------------------------------------------------------------
// GEMM2: mu_rawT(K x C) = zsw^T (K x HW) * xT (HW x C); contraction HW=4096
// ---------------------------------------------------------------------------
__global__ void gemm2_kernel(const float* __restrict__ zsw,
                             const float* __restrict__ xT,
                             float* __restrict__ mu_rawT) {
  const int lane = threadIdx.x & 31;
  const int l    = lane & 15;
  const int half = lane >> 4;
  const int b    = blockIdx.y;
  const int k0   = (blockIdx.x >> 2) * 16;
  const int c0b  = (blockIdx.x & 3) * 64;
  const float* zb = zsw + (size_t)b * HW * K_DIM;
  const float* xb = xT + (size_t)b * HW * C_DIM;

  v8f acc[4];
#pragma unroll
  for (int t = 0; t < 4; ++t) acc[t] = vzero8();

  for (int n0 = 0; n0 < HW; n0 += 4) {
    v2f a;
    a.x = zb[(size_t)(n0 + 2 * half) * K_DIM + k0 + l];
    a.y = zb[(size_t)(n0 + 2 * half + 1) * K_DIM + k0 + l];
    const float* bp = xb + (size_t)(n0 + 2 * half) * C_DIM + c0b + l;
#pragma unroll
    for (int t = 0; t < 4; ++t) {
      v2f bf;
      bf.x = bp[t * 16];
      bf.y = bp[C_DIM + t * 16];
      acc[t] = wmma_f32(a, bf, acc[t]);
    }
  }
  float* ob = mu_rawT + (size_t)b * K_DIM * C_DIM;
#pragma unroll
  for (int t = 0; t < 4; ++t)
#pragma unroll
    for (int r = 0; r < 8; ++r)
      ob[(size_t)(k0 + r + 8 * half) * C_DIM + c0b + t * 16 + l] = acc[t][r];
}

// ---------------------------------------------------------------------------
// mu update: scale column by 1/(1e-6+colsum), then l2norm over c (with 1e-6)
// writes mu (C,K), muT (K,C) and the mu output region of d_out
// ---------------------------------------------------------------------------
__global__ void mu_norm_kernel(const float* __restrict__ mu_rawT,
                               const float* __restrict__ colsum,
                               float* __restrict__ mu, float* __restrict__ muT,
                               float* __restrict__ mu_out) {
  __shared__ float sm[64];
  int k = blockIdx.x, b = blockIdx.y, t = threadIdx.x;
  const float* rowp = mu_rawT + ((size_t)b * K_DIM + k) * C_DIM;
  float invden = 1.0f / (1e-6f + colsum[b * K_DIM + k]);
  float v[4]; float ss = 0.0f;
#pragma unroll
  for (int j = 0; j < 4; ++j) {
    v[j] = rowp[t + 64 * j] * invden;
    ss += v[j] * v[j];
  }
  sm[t] = ss; __syncthreads();
  for (int off = 32; off; off >>= 1) {
    if (t < off) sm[t] += sm[t + off];
    __syncthreads();
  }
  float rn = 1.0f / (1e-6f + sqrtf(sm[0]));
#pragma unroll
  for (int j = 0; j < 4; ++j) {
    int c = t + 64 * j;
    float val = v[j] * rn;
    mu[((size_t)b * C_DIM + c) * K_DIM + k]     = val;
    muT[((size_t)b * K_DIM + k) * C_DIM + c]    = val;
    mu_out[((size_t)b * C_DIM + c) * K_DIM + k] = val;
  }
}

// ---------------------------------------------------------------------------
// feat: featT[b][si][yx][c] = sum_k zsw[yx][k] * muT[k][c]  (fold collapsed)
// ---------------------------------------------------------------------------
__global__ void feat_kernel(const float* __restrict__ zsw,
                            const float* __restrict__ muT,
                            float* __restrict__ featT, int si) {
  const int lane = threadIdx.x & 31;
  const int l    = lane & 15;
  const int half = lane >> 4;
  const int b    = blockIdx.y;
  const int row0 = (blockIdx.x >> 2) * 16;
  const int c0b  = (blockIdx.x & 3) * 64;
  const float* zb = zsw + (size_t)b * HW * K_DIM;
  const float* mb = muT + (size_t)b * K_DIM * C_DIM;

  v8f acc[4];
#pragma unroll
  for (int t = 0; t < 4; ++t) acc[t] = vzero8();

  const float* arow = zb + (size_t)(row0 + l) * K_DIM + 2 * half;
  for (int k0 = 0; k0 < K_DIM; k0 += 4) {
    v2f a = *(const v2f*)(arow + k0);
    const float* bp = mb + (size_t)(k0 + 2 * half) * C_DIM + c0b + l;
#pragma unroll
    for (int t = 0; t < 4; ++t) {
      v2f bf;
      bf.x = bp[t * 16];
      bf.y = bp[C_DIM + t * 16];
      acc[t] = wmma_f32(a, bf, acc[t]);
    }
  }
  float* ob = featT + (((size_t)b * 3 + si) * HW) * C_DIM;
#pragma unroll
  for (int t = 0; t < 4; ++t)
#pragma unroll
    for (int r = 0; r < 8; ++r)
      ob[(size_t)(row0 + r + 8 * half) * C_DIM + c0b + t * 16 + l] = acc[t][r];
}

// ---------------------------------------------------------------------------
// 1x1 conv over concatenated feats (contraction 768) + bias, *w, +x, ReLU
// ---------------------------------------------------------------------------
__global__ void conv_out_kernel(const float* __restrict__ featT,
                                const float* __restrict__ w_catT,
                                const float* __restrict__ b_cat,
                                const float* __restrict__ w_scl,
                                const float* __restrict__ x,
                                float* __restrict__ out) {
  const int lane = threadIdx.x & 31;
  const int l    = lane & 15;
  const int half = lane >> 4;
  const int b    = blockIdx.y;
  const int row0 = (blockIdx.x >> 2) * 16;
  const int o0b  = (blockIdx.x & 3) * 64;
  const float* fb = featT + (size_t)b * 3 * HW * C_DIM;

  v8f acc[4];
#pragma unroll
  for (int t = 0; t < 4; ++t) acc[t] = vzero8();

  for (int i0 = 0; i0 < W3C; i0 += 4) {
    int i    = i0 + 2 * half;
    int sidx = i >> 8;
    int c    = i & 255;
    v2f a = *(const v2f*)(fb + ((size_t)sidx * HW + row0 + l) * C_DIM + c);
    const float* bp = w_catT + (size_t)i * C_DIM + o0b + l;
#pragma unroll
    for (int t = 0; t < 4; ++t) {
      v2f bf;
      bf.x = bp[t * 16];
      bf.y = bp[C_DIM + t * 16];
      acc[t] = wmma_f32(a, bf, acc[t]);
    }
  }
  float w = w_scl[0];
#pragma unroll
  for (int t = 0; t < 4; ++t) {
    int o = o0b + t * 16 + l;
    float bc = b_cat[o];
#pragma unroll
    for (int r = 0; r < 8; ++r) {
      int yx = row0 + r + 8 * half;
      size_t idx = ((size_t)b * C_DIM + o) * HW + yx;
      float u = (acc[t][r] + bc) * w + x[idx];
      out[idx] = fmaxf(u, 0.0f);
    }
  }
}

// ---------------------------------------------------------------------------
extern "C" void kernel_launch(void* const* d_in, const int* in_sizes, int n_in,
                              void* d_out, int out_size, void* d_ws,
                              size_t ws_size, hipStream_t stream) {
  (void)in_sizes; (void)n_in; (void)out_size; (void)ws_size;
  const float* x    = (const float*)d_in[0];
  const float* mu0  = (const float*)d_in[1];
  const float* wscl = (const float*)d_in[2];
  const float* wcat = (const float*)d_in[3];
  const float* bcat = (const float*)d_in[4];

  float* out    = (float*)d_out;
  float* mu_out = out + (size_t)B_DIM * C_DIM * HW;  // (b, C, K) tail

  float* ws = (float*)d_ws;
  size_t off = 0;
  float* xT      = ws + off; off += (size_t)B_DIM * HW * C_DIM;     // 4.19M
  float* w_catT  = ws + off; off += (size_t)W3C * C_DIM;            // 196K
  float* mu      = ws + off; off += (size_t)B_DIM * C_DIM * K_DIM;  // 65K
  float* muT     = ws + off; off += (size_t)B_DIM * K_DIM * C_DIM;  // 65K
  float* zsw     = ws + off; off += (size_t)B_DIM * HW * K_DIM;     // 1.05M
  float* colsum  = ws + off; off += (size_t)B_DIM * K_DIM;          // 256
  float* mu_rawT = ws + off; off += (size_t)B_DIM * K_DIM * C_DIM;  // 65K
  float* featT   = ws + off;                                        // 12.6M

  {
    int total = B_DIM * HW * C_DIM;
    prep_kernel<<<(total + 255) / 256, 256, 0, stream>>>(
        x, mu0, wcat, xT, w_catT, mu, muT);
  }

  for (int si = 0; si < 3; ++si) {
    int s = 2 * si + 1;
    int p = si;
    long sc = 0;
    for (int t = 0; t < 64; ++t) {
      int lo = t + p - 63; if (lo < 0) lo = 0;
      int hi = t + p;      if (hi > s - 1) hi = s - 1;
      sc += (hi - lo + 1);
    }
    float oobterm = (float)((long)s * s * HW - sc * sc) / (float)K_DIM;

    for (int stage = 0; stage < 3; ++stage) {
      gemm1_softmax_kernel<<<dim3(HW / 16, B_DIM), 32, 0, stream>>>(
          xT, mu, zsw, s, p);
      colsum_kernel<<<dim3(K_DIM, B_DIM), 256, 0, stream>>>(
          zsw, colsum, oobterm);
      gemm2_kernel<<<dim3(16, B_DIM), 32, 0, stream>>>(zsw, xT, mu_rawT);
      mu_norm_kernel<<<dim3(K_DIM, B_DIM), 64, 0, stream>>>(
          mu_rawT, colsum, mu, muT, mu_out);
    }
    feat_kernel<<<dim3((HW / 16) * 4, B_DIM), 32, 0, stream>>>(
        zsw, muT, featT, si);
  }

  conv_out_kernel<<<dim3((HW / 16) * 4, B_DIM), 32, 0, stream>>>(
      featT, w_catT, bcat, wscl, x, out);
}